// NerfRender_26731876450381
// MI455X (gfx1250) — compile-verified
//
#include <hip/hip_runtime.h>

typedef __attribute__((ext_vector_type(2))) float v2f;
typedef __attribute__((ext_vector_type(8))) float v8f;

#define N_SAMP 256
#define EPSV 1e-5f

__global__ __launch_bounds__(512) void nerf_render_kernel(
    const float* __restrict__ t, const float* __restrict__ sdf,
    const float* __restrict__ color, const float* __restrict__ s_inv_log,
    float* __restrict__ c_out, float* __restrict__ d_out,
    float* __restrict__ wi_out, float* __restrict__ t_out, int n_rays)
{
    __shared__ float lds_part[16][4][4];   // [ray-in-block][quantity rgb,t][4 partials]

    const int lane = threadIdx.x & 31;
    const int wave = threadIdx.x >> 5;
    const int ray  = blockIdx.x * 16 + wave;

    const float s = expf(-s_inv_log[0]);

    const float* sdf_ray = sdf   + (size_t)ray * N_SAMP;
    const float* t_ray   = t     + (size_t)ray * N_SAMP;
    const float* col_ray = color + (size_t)ray * N_SAMP * 3;

    // q = sigmoid(s * sdf) at all 256 samples; qa[c] = q at sample 32c+lane.
    // Lane l <-> sample 32c+l keeps every global access unit-stride per wave.
    float qa[8];
#pragma unroll
    for (int c = 0; c < 8; ++c) {
        float x = s * sdf_ray[(c << 5) + lane];
        qa[c] = 1.0f / (1.0f + expf(-x));
    }

    float Tc = 1.0f;                 // transmittance carried across chunks
    float cr = 0.0f, cg = 0.0f, cb = 0.0f, dd = 0.0f;

#pragma unroll
    for (int c = 0; c < 8; ++c) {
        const int i = (c << 5) + lane;
        float q  = qa[c];
        float qn = __shfl_down(q, 1, 32);
        float fn = __shfl(qa[c < 7 ? c + 1 : 7], 0, 32);   // lane 0 of next chunk
        if (lane == 31) qn = fn;

        float a = (q - qn + EPSV) / (q + EPSV);
        a = fminf(fmaxf(a, 0.0f), 1.0f);
        if (c == 7 && lane == 31) a = 0.0f;   // interval 255 does not exist
        float om = 1.0f - a;

        // exclusive multiplicative Hillis-Steele scan of om across the wave
        float x = __shfl_up(om, 1, 32);
        if (lane == 0) x = 1.0f;
#pragma unroll
        for (int off = 1; off < 32; off <<= 1) {
            float y = __shfl_up(x, off, 32);
            if (lane >= off) x *= y;
        }
        float Ti = Tc * x;          // T at sample i
        float wi = Ti * a;
        Tc *= __shfl(x * om, 31, 32);   // fold chunk's total product into carry

        float tv = t_ray[i];
        float c0 = col_ray[3 * i + 0];
        float c1 = col_ray[3 * i + 1];
        float c2 = col_ray[3 * i + 2];
        cr += wi * c0; cg += wi * c1; cb += wi * c2; dd += wi * tv;

        if (i < 255) {
            wi_out[(size_t)ray * 255 + i] = wi;   // coalesced: addr = lane-stride 1
            t_out [(size_t)ray * 255 + i] = tv;
        }
    }

    // Reduce 32 lane-partials -> 4 partials per quantity (lanes 0..3 hold them)
#pragma unroll
    for (int m = 16; m >= 4; m >>= 1) {
        cr += __shfl_xor(cr, m, 32);
        cg += __shfl_xor(cg, m, 32);
        cb += __shfl_xor(cb, m, 32);
        dd += __shfl_xor(dd, m, 32);
    }
    if (lane < 4) {
        lds_part[wave][0][lane] = cr;
        lds_part[wave][1][lane] = cg;
        lds_part[wave][2][lane] = cb;
        lds_part[wave][3][lane] = dd;
    }
    __syncthreads();

    // Wave 0: 16-ray x 4-partial row-sum via V_WMMA_F32_16X16X4_F32 with B=ones.
    // A layout (documented): lanes 0-15 supply K=0,1 ; lanes 16-31 supply K=2,3.
    // B=ones makes the f32 B layout irrelevant. D: VGPR v, lane<16 -> M=v,
    // lane>=16 -> M=v+8; every column holds the row sum.
    if (wave == 0) {
        const int row = lane & 15;
        const int kb  = (lane >> 4) << 1;
        v2f bmat = {1.0f, 1.0f};
#pragma unroll
        for (int q = 0; q < 4; ++q) {
            v2f amat = { lds_part[row][q][kb], lds_part[row][q][kb + 1] };
            v8f cacc = {};
            v8f dres = __builtin_amdgcn_wmma_f32_16x16x4_f32(
                false, amat, false, bmat, (short)0, cacc, false, false);
            if (row == 0) {                    // lanes 0 and 16 write results
                int rbase = blockIdx.x * 16 + ((lane >> 4) << 3);
#pragma unroll
                for (int v = 0; v < 8; ++v) {
                    float val = dres[v];
                    int rr = rbase + v;
                    if (q < 3) c_out[(size_t)rr * 3 + q] = val;
                    else       d_out[rr] = val;
                }
            }
        }
    }
}

extern "C" void kernel_launch(void* const* d_in, const int* in_sizes, int n_in,
                              void* d_out, int out_size, void* d_ws, size_t ws_size,
                              hipStream_t stream) {
    const float* t     = (const float*)d_in[0];
    const float* sdf   = (const float*)d_in[1];
    const float* color = (const float*)d_in[2];
    const float* sil   = (const float*)d_in[3];

    const int R = in_sizes[0] / N_SAMP;   // 65536

    float* out    = (float*)d_out;
    float* c_out  = out;                               // R*3
    float* dpt    = out + 3 * (size_t)R;               // R
    float* wi     = out + 4 * (size_t)R;               // R*255
    float* tout   = wi  + 255 * (size_t)R;             // R*255

    dim3 grid(R / 16), block(512);
    hipLaunchKernelGGL(nerf_render_kernel, grid, block, 0, stream,
                       t, sdf, color, sil, c_out, dpt, wi, tout, R);
}